// GGNN_46033459479017
// MI455X (gfx1250) — compile-verified
//
#include <hip/hip_runtime.h>
#include <hip/hip_bf16.h>

// ---------------- problem constants ----------------
#define NN 1024
#define HH 96
#define EE 16384
#define H3 288

typedef __attribute__((ext_vector_type(2))) float v2f;
typedef __attribute__((ext_vector_type(8))) float v8f;

__device__ __forceinline__ float sigf(float v) {
    return 1.0f / (1.0f + __expf(-v));
}
__device__ __forceinline__ float leakyf(float v) {
    return (v >= 0.0f) ? v : 0.01f * v;
}

// ---------------- init: x = leaky(x_raw[:, :9] @ W_init.T + b_init) ----------------
__global__ void k_init(const float* __restrict__ xr, const float* __restrict__ Wi,
                       const float* __restrict__ bi, float* __restrict__ x) {
    int i = blockIdx.x * blockDim.x + threadIdx.x;     // 0 .. N*H-1
    if (i >= NN * HH) return;
    int n = i / HH, h = i % HH;
    float acc = bi[h];
    #pragma unroll
    for (int k = 0; k < 9; ++k)
        acc += xr[n * 10 + k] * Wi[h * 9 + k];
    x[i] = leakyf(acc);
}

// ---------------- generic fp32 WMMA GEMM: C[M x Nc] = A[M x K] * B ----------------
// 4 waves per block, one 16x16 output tile per wave.
// grid = (Nc/16, M/64), block = (32,4).  K must be a multiple of 16.
// BTRANS==0: B row-major [K x Nc]   (elem B[k*ldb + n])
// BTRANS==1: B stored  [Nc x K]     (elem B[n*ldb + k], i.e. multiply by stored^T)
// ACT: 0 = none, 1 = leaky-relu.  HASBIAS: per-output-column bias.
template <int BTRANS, int ACT, bool HASBIAS>
__global__ void k_gemm_wmma(const float* __restrict__ A, int lda,
                            const float* __restrict__ B, int ldb,
                            const float* __restrict__ bias,
                            float* __restrict__ C, int ldc, int K) {
    const int lane = threadIdx.x;                       // 0..31
    const int half = lane >> 4;                         // 0|1
    const int l16  = lane & 15;
    const int col0 = blockIdx.x * 16;
    const int row0 = (blockIdx.y * 4 + threadIdx.y) * 16;
    const int arow = row0 + l16;
    const int bcol = col0 + l16;

    // per-lane base pointers: K offsets {k+2h, k+2h+1} per ISA 16x4 f32 A layout
    const float* __restrict__ Ap = A + (size_t)arow * lda + 2 * half;
    const float* __restrict__ Bt = B + (size_t)bcol * ldb + 2 * half;   // BTRANS path
    const float* __restrict__ Bn = B + (size_t)(2 * half) * ldb + bcol; // non-trans path

    v8f acc = {};
    for (int k = 0; k < K; k += 16) {
        v2f a[4], b[4];
        #pragma unroll
        for (int u = 0; u < 4; ++u) {
            const int ko = k + 4 * u;
            a[u] = *(const v2f*)(Ap + ko);
            if (BTRANS) {
                b[u] = *(const v2f*)(Bt + ko);
            } else {
                b[u].x = Bn[(size_t)ko * ldb];
                b[u].y = Bn[(size_t)(ko + 1) * ldb];
            }
        }
        #pragma unroll
        for (int u = 0; u < 4; ++u)
            acc = __builtin_amdgcn_wmma_f32_16x16x4_f32(
                      false, a[u], false, b[u], (short)0, acc, false, false);
    }

    const float bv = HASBIAS ? bias[bcol] : 0.0f;
    #pragma unroll
    for (int r = 0; r < 8; ++r) {
        const int row = row0 + r + 8 * half;   // C/D layout: lanes16-31 hold M=r+8
        float v = acc[r] + bv;
        if (ACT == 1) v = leakyf(v);
        C[(size_t)row * ldc + bcol] = v;
    }
}

// ---------------- zero scratch ----------------
__global__ void k_zero(float* __restrict__ p, int n) {
    int i = blockIdx.x * blockDim.x + threadIdx.x;
    if (i < n) p[i] = 0.0f;
}

// ---------------- edge scatter: s[tgt] += m[src], cnt[tgt] += 1 ----------------
__global__ void k_scatter(const long long* __restrict__ ei, const float* __restrict__ m,
                          float* __restrict__ s, float* __restrict__ cnt) {
    int e = blockIdx.x;            // 0..E-1
    int f = threadIdx.x;           // 0..95
    int sn = (int)ei[e];
    int tn = (int)ei[EE + e];
    atomicAdd(&s[tn * HH + f], m[sn * HH + f]);
    if (f == 0) atomicAdd(&cnt[tn], 1.0f);
}

// ---------------- agg = s / max(cnt,1) (in place) ----------------
__global__ void k_div(float* __restrict__ s, const float* __restrict__ cnt) {
    int i = blockIdx.x * blockDim.x + threadIdx.x;
    if (i >= NN * HH) return;
    s[i] = s[i] / fmaxf(cnt[i / HH], 1.0f);
}

// ---------------- GRU gate fusion ----------------
__global__ void k_gru(const float* __restrict__ gi, const float* __restrict__ gh,
                      const float* __restrict__ x, float* __restrict__ xg) {
    int i = blockIdx.x * blockDim.x + threadIdx.x;
    if (i >= NN * HH) return;
    int n = i / HH, h = i % HH;
    const float* gin = gi + n * H3;
    const float* ghn = gh + n * H3;
    float r  = sigf(gin[h]            + ghn[h]);
    float z  = sigf(gin[HH + h]       + ghn[HH + h]);
    float nn = tanhf(gin[2 * HH + h] + r * ghn[2 * HH + h]);
    xg[i] = (1.0f - z) * nn + z * x[i];
}

// ---------------- transpose hj [N,H] -> hjT [H,N] for coalesced attn loads ----------------
__global__ void k_transpose(const float* __restrict__ hj, float* __restrict__ hjT) {
    int i = blockIdx.x * blockDim.x + threadIdx.x;
    if (i >= NN * HH) return;
    int n = i / HH, k = i % HH;
    hjT[k * NN + n] = hj[i];
}

// ---------------- attention coefficients ----------------
// coeffs[i,j] = adj[i,j]==1 ? sigmoid( sum_k sigmoid(hi[i,k]+hj[j,k]+ba1[k]) * Wa2[k] + ba2 ) : 0
__global__ void k_attn(const float* __restrict__ hi, const float* __restrict__ hjT,
                       const float* __restrict__ ba1, const float* __restrict__ Wa2,
                       const float* __restrict__ ba2, const int* __restrict__ adj,
                       float* __restrict__ co) {
    __shared__ float sh[HH];
    __shared__ float sw[HH];
    const int i = blockIdx.x;
    const int j = blockIdx.y * blockDim.x + threadIdx.x;
    if (threadIdx.x < HH) {
        sh[threadIdx.x] = hi[i * HH + threadIdx.x] + ba1[threadIdx.x];
        sw[threadIdx.x] = Wa2[threadIdx.x];
    }
    __syncthreads();
    float acc = ba2[0];
    #pragma unroll 8
    for (int k = 0; k < HH; ++k) {
        float v = sh[k] + hjT[k * NN + j];
        acc += sigf(v) * sw[k];
    }
    float c = sigf(acc);
    co[i * NN + j] = (adj[i * NN + j] == 1) ? c : 0.0f;
}

// ---------------- output heads ----------------
__global__ void k_head(const float* __restrict__ xo,
                       const float* __restrict__ Wp1, const float* __restrict__ bp1,
                       const float* __restrict__ Wp2, const float* __restrict__ bp2,
                       float* __restrict__ out) {
    int n = blockIdx.x * blockDim.x + threadIdx.x;
    if (n >= NN) return;
    float a1 = bp1[0], a2 = bp2[0];
    #pragma unroll 8
    for (int h = 0; h < HH; ++h) {
        float v = xo[n * HH + h];
        a1 += v * Wp1[h];
        a2 += v * Wp2[h];
    }
    out[n]      = sigf(a1);
    out[NN + n] = sigf(a2);
}

// ---------------- host orchestration ----------------
extern "C" void kernel_launch(void* const* d_in, const int* in_sizes, int n_in,
                              void* d_out, int out_size, void* d_ws, size_t ws_size,
                              hipStream_t stream) {
    const float*     x_raw = (const float*)d_in[0];
    const long long* ei    = (const long long*)d_in[1];   // int64 edge_index [2,E]
    const int*       adj   = (const int*)d_in[2];
    const float* W_init = (const float*)d_in[3];
    const float* b_init = (const float*)d_in[4];
    const float* W_ggc  = (const float*)d_in[5];
    const float* W_ih   = (const float*)d_in[6];
    const float* W_hh   = (const float*)d_in[7];
    const float* b_ih   = (const float*)d_in[8];
    const float* b_hh   = (const float*)d_in[9];
    const float* Wa1    = (const float*)d_in[10];
    const float* ba1    = (const float*)d_in[11];
    const float* Wa2    = (const float*)d_in[12];
    const float* ba2    = (const float*)d_in[13];
    const float* Wo1    = (const float*)d_in[14];
    const float* bo1    = (const float*)d_in[15];
    const float* Wo2    = (const float*)d_in[16];
    const float* bo2    = (const float*)d_in[17];
    const float* Wp1    = (const float*)d_in[18];
    const float* bp1    = (const float*)d_in[19];
    const float* Wp2    = (const float*)d_in[20];
    const float* bp2    = (const float*)d_in[21];
    float* out = (float*)d_out;

    // workspace carve-up (floats)
    float* ws  = (float*)d_ws;
    float* XB  = ws;                      // x            [N,H]
    float* XG  = XB  + NN * HH;           // x after GRU  [N,H]
    float* M   = XG  + NN * HH;           // messages / xo1
    float* AGG = M   + NN * HH;           // s -> agg
    float* CNT = AGG + NN * HH;           // [N]
    float* GI  = CNT + NN;                // [N,3H]
    float* GH  = GI  + NN * H3;           // [N,3H]
    float* HI  = GH  + NN * H3;           // [N,H] / xo2
    float* HJ  = HI  + NN * HH;           // [N,H]
    float* HJT = HJ  + NN * HH;           // [H,N]
    float* CO  = HJT + NN * HH;           // coeffs [N,N]

    const int NH = NN * HH;               // 98304
    const dim3 b256(256);
    const dim3 gNH((NH + 255) / 256);
    const dim3 wblk(32, 4);               // 4 waves/block, 1 tile/wave
    const dim3 gH (HH / 16, NN / 64);     // (6,16)  -> [N,H]  tiles
    const dim3 g3H(H3 / 16, NN / 64);     // (18,16) -> [N,3H] tiles

    // x0 = leaky(x[:, :9] @ W_init.T + b_init)
    k_init<<<gNH, b256, 0, stream>>>(x_raw, W_init, b_init, XB);

    for (int t = 0; t < 2; ++t) {
        // m = x @ W_ggc
        k_gemm_wmma<0, 0, false><<<gH, wblk, 0, stream>>>(XB, HH, W_ggc, HH, nullptr, M, HH, HH);
        // segment mean aggregation
        k_zero<<<gNH, b256, 0, stream>>>(AGG, NH);
        k_zero<<<dim3((NN + 255) / 256), b256, 0, stream>>>(CNT, NN);
        k_scatter<<<dim3(EE), dim3(HH), 0, stream>>>(ei, M, AGG, CNT);
        k_div<<<gNH, b256, 0, stream>>>(AGG, CNT);
        // gi = agg @ W_ih.T + b_ih ; gh = x @ W_hh.T + b_hh
        k_gemm_wmma<1, 0, true><<<g3H, wblk, 0, stream>>>(AGG, HH, W_ih, HH, b_ih, GI, H3, HH);
        k_gemm_wmma<1, 0, true><<<g3H, wblk, 0, stream>>>(XB,  HH, W_hh, HH, b_hh, GH, H3, HH);
        // GRU update -> XG
        k_gru<<<gNH, b256, 0, stream>>>(GI, GH, XB, XG);
        // hi = xg @ Wa1[:, :H].T ; hj = xg @ Wa1[:, H:].T   (Wa1 is [H, 2H])
        k_gemm_wmma<1, 0, false><<<gH, wblk, 0, stream>>>(XG, HH, Wa1,      2 * HH, nullptr, HI, HH, HH);
        k_gemm_wmma<1, 0, false><<<gH, wblk, 0, stream>>>(XG, HH, Wa1 + HH, 2 * HH, nullptr, HJ, HH, HH);
        k_transpose<<<gNH, b256, 0, stream>>>(HJ, HJT);
        // attention coefficients (masked)
        k_attn<<<dim3(NN, NN / 256), b256, 0, stream>>>(HI, HJT, ba1, Wa2, ba2, adj, CO);
        // x = coeffs @ xg   (K = 1024) -> back into XB for next iteration
        k_gemm_wmma<0, 0, false><<<gH, wblk, 0, stream>>>(CO, NN, XG, HH, nullptr, XB, HH, NN);
    }

    // output MLP: xo1 = leaky(x@Wo1.T+bo1) ; xo2 = leaky(xo1@Wo2.T+bo2)
    k_gemm_wmma<1, 1, true><<<gH, wblk, 0, stream>>>(XB, HH, Wo1, HH, bo1, M,  HH, HH);
    k_gemm_wmma<1, 1, true><<<gH, wblk, 0, stream>>>(M,  HH, Wo2, HH, bo2, HI, HH, HH);
    // heads -> d_out = [out1 (N) | out2 (N)]
    k_head<<<dim3(NN / 256), b256, 0, stream>>>(HI, Wp1, bp1, Wp2, bp2, out);
    (void)in_sizes; (void)n_in; (void)out_size; (void)ws_size;
}